// FCOSPostProcessor_81243601371357
// MI455X (gfx1250) — compile-verified
//
#include <hip/hip_runtime.h>
#include <hip/hip_bf16.h>
#include <stdint.h>

// ---------------- problem constants ----------------
#define NIMG  16
#define NCLS  80
#define HH    160
#define WW    160
#define LL    (HH*WW)        // 25600 locations per image
#define NBINS 4096
#define KTOP  1000
#define CAP   4096           // per-image collect buffer capacity
#define SORTM 4096           // bitonic sort size (power of two >= CAP)
#define CAND_THRESH (-2.9444389791664403f)   // logit(0.05)

// ---------------- workspace layout (bytes) ----------------
// [0,        524288)  : u64 buf[NIMG][CAP]
// [524288,  2162688)  : f32 ctrsig[NIMG*LL]
// [2162688, 2424832)  : u32 hist[NIMG][NBINS]
// [2424832, 2424896)  : u32 Tbin[NIMG]
// [2424896, 2424960)  : u32 cnt[NIMG]

__device__ __forceinline__ float sigmoidf_(float x) {
    return 1.0f / (1.0f + __expf(-x));
}

// low 32 bits of a generic pointer to LDS == wave-relative LDS byte offset
__device__ __forceinline__ unsigned lds_off_u32(const void* p) {
    return (unsigned)(unsigned long long)p;
}

// CDNA5 async global->LDS copy, 16B per lane (ASYNCcnt-tracked)
__device__ __forceinline__ void async_copy_b128(unsigned ldsoff, const void* gaddr) {
    asm volatile("global_load_async_to_lds_b128 %0, %1, off"
                 :: "v"(ldsoff), "v"((unsigned long long)gaddr)
                 : "memory");
}
__device__ __forceinline__ void wait_async_le1() {
    asm volatile("s_wait_asynccnt 0x1" ::: "memory");
}
__device__ __forceinline__ void wait_async_le0() {
    asm volatile("s_wait_asynccnt 0x0" ::: "memory");
}

// ---------------- init: zero histograms & counters ----------------
__global__ void k_init(unsigned* __restrict__ hist, unsigned* __restrict__ cnt) {
    int t = blockIdx.x * 256 + threadIdx.x;
    if (t < NIMG * NBINS) hist[t] = 0u;
    if (t < NIMG)         cnt[t]  = 0u;
}

// ---------------- pass 0: sigmoid(centerness) ----------------
__global__ void k_ctr(const float* __restrict__ ctr, float* __restrict__ ctrsig, int n) {
    int t = blockIdx.x * 256 + threadIdx.x;
    if (t < n) ctrsig[t] = sigmoidf_(ctr[t]);
}

// ---------------- pass 1: per-image score histogram ----------------
// one block per (image,class) plane; async double-buffered streaming of cls
__global__ void __launch_bounds__(256) k_hist(const float* __restrict__ cls,
                                              const float* __restrict__ ctrsig,
                                              unsigned* __restrict__ hist) {
    __shared__ unsigned h[NBINS];
    __shared__ __align__(16) float tile[2][1024];

    const int tid   = threadIdx.x;
    const int plane = blockIdx.x;          // img*NCLS + c
    const int img   = plane / NCLS;
    const float* src  = cls    + (size_t)plane * LL;
    const float* csig = ctrsig + (size_t)img   * LL;

    for (int j = tid; j < NBINS; j += 256) h[j] = 0u;
    __syncthreads();

    const unsigned lds0 = lds_off_u32(&tile[0][tid * 4]);
    const unsigned lds1 = lds_off_u32(&tile[1][tid * 4]);
    const int ITER = LL / 1024;            // 25, exact

    async_copy_b128(lds0, src + tid * 4);
    for (int it = 0; it < ITER; ++it) {
        if (it + 1 < ITER) {
            async_copy_b128((it & 1) ? lds0 : lds1, src + (it + 1) * 1024 + tid * 4);
            wait_async_le1();
        } else {
            wait_async_le0();
        }
        float4 v = *reinterpret_cast<const float4*>(&tile[it & 1][tid * 4]);
        const int loc = it * 1024 + tid * 4;
        float xs[4] = {v.x, v.y, v.z, v.w};
#pragma unroll
        for (int q = 0; q < 4; ++q) {
            float x = xs[q];
            if (x > CAND_THRESH) {
                float s = sigmoidf_(x) * csig[loc + q];
                int b = (int)(s * (float)NBINS);
                b = b > (NBINS - 1) ? (NBINS - 1) : b;
                atomicAdd(&h[b], 1u);
            }
        }
    }
    __syncthreads();
    unsigned* gh = hist + (size_t)img * NBINS;
    for (int j = tid; j < NBINS; j += 256) {
        unsigned c = h[j];
        if (c) atomicAdd(&gh[j], c);
    }
}

// ---------------- pass 2: find threshold bin per image ----------------
__global__ void k_thresh(const unsigned* __restrict__ hist, unsigned* __restrict__ Tbin) {
    int i = threadIdx.x;
    if (i < NIMG) {
        const unsigned* h = hist + (size_t)i * NBINS;
        unsigned cum = 0, T = 0;
        for (int b = NBINS - 1; b >= 0; --b) {
            cum += h[b];
            if (cum >= KTOP) { T = (unsigned)b; break; }
        }
        Tbin[i] = T;
    }
}

// ---------------- pass 3: collect keys with bin >= T ----------------
__global__ void __launch_bounds__(256) k_collect(const float* __restrict__ cls,
                                                 const float* __restrict__ ctrsig,
                                                 const unsigned* __restrict__ Tbin,
                                                 unsigned* __restrict__ cnt,
                                                 unsigned long long* __restrict__ buf) {
    __shared__ __align__(16) float tile[2][1024];

    const int tid   = threadIdx.x;
    const int plane = blockIdx.x;
    const int img   = plane / NCLS;
    const int c     = plane - img * NCLS;
    const float* src  = cls    + (size_t)plane * LL;
    const float* csig = ctrsig + (size_t)img   * LL;
    const unsigned T  = Tbin[img];

    const unsigned lds0 = lds_off_u32(&tile[0][tid * 4]);
    const unsigned lds1 = lds_off_u32(&tile[1][tid * 4]);
    const int ITER = LL / 1024;

    async_copy_b128(lds0, src + tid * 4);
    for (int it = 0; it < ITER; ++it) {
        if (it + 1 < ITER) {
            async_copy_b128((it & 1) ? lds0 : lds1, src + (it + 1) * 1024 + tid * 4);
            wait_async_le1();
        } else {
            wait_async_le0();
        }
        float4 v = *reinterpret_cast<const float4*>(&tile[it & 1][tid * 4]);
        const int loc0 = it * 1024 + tid * 4;
        float xs[4] = {v.x, v.y, v.z, v.w};
#pragma unroll
        for (int q = 0; q < 4; ++q) {
            float x = xs[q];
            if (x > CAND_THRESH) {
                float s = sigmoidf_(x) * csig[loc0 + q];
                int b = (int)(s * (float)NBINS);
                b = b > (NBINS - 1) ? (NBINS - 1) : b;
                if ((unsigned)b >= T) {
                    unsigned idxf = (unsigned)((loc0 + q) * NCLS + c);  // < 2^21
                    unsigned long long key =
                        ((unsigned long long)__float_as_uint(s) << 21) |
                        (unsigned long long)(0x1FFFFFu - idxf);
                    unsigned pos = atomicAdd(&cnt[img], 1u);
                    if (pos < CAP) buf[(size_t)img * CAP + pos] = key;
                }
            }
        }
    }
}

// ---------------- pass 4: per-image bitonic sort + decode/output ----------------
__global__ void __launch_bounds__(1024) k_final(const unsigned* __restrict__ cnt,
                                                const unsigned long long* __restrict__ buf,
                                                const float* __restrict__ reg,
                                                const int* __restrict__ imsz,
                                                float* __restrict__ out) {
    __shared__ unsigned long long key[SORTM];
    const int i   = blockIdx.x;
    const int tid = threadIdx.x;
    const int bdim = blockDim.x;

    unsigned n = cnt[i];
    if (n > CAP) n = CAP;
    for (int j = tid; j < SORTM; j += bdim)
        key[j] = (j < (int)n) ? buf[(size_t)i * CAP + j] : 0ULL;
    __syncthreads();

    // descending bitonic sort
    for (int kk = 2; kk <= SORTM; kk <<= 1) {
        for (int jj = kk >> 1; jj > 0; jj >>= 1) {
            for (int t = tid; t < SORTM; t += bdim) {
                int ixj = t ^ jj;
                if (ixj > t) {
                    unsigned long long a = key[t], b = key[ixj];
                    bool descBlock = ((t & kk) == 0);
                    bool sw = descBlock ? (a < b) : (a > b);
                    if (sw) { key[t] = b; key[ixj] = a; }
                }
            }
            __syncthreads();
        }
    }

    const float wmax = (float)imsz[i * 2 + 1] - 1.0f;   // TO_REMOVE = 1
    const float hmax = (float)imsz[i * 2 + 0] - 1.0f;

    for (int j = tid; j < KTOP; j += bdim) {
        unsigned long long kj = key[j];
        float x1 = 0.f, y1 = 0.f, x2 = 0.f, y2 = 0.f;
        float sc = 0.f, lab = 1.0f, val = 0.f;
        if (kj != 0ULL) {
            unsigned idxf = 0x1FFFFFu - (unsigned)(kj & 0x1FFFFFull);
            float s = __uint_as_float((unsigned)(kj >> 21));
            int loc = (int)(idxf / NCLS);
            int c   = (int)idxf - loc * NCLS;
            lab = (float)(c + 1);
            float lx = ((float)(loc % WW) + 0.5f) * 8.0f;
            float ly = ((float)(loc / WW) + 0.5f) * 8.0f;
            const float* rg = reg + (size_t)i * 4 * LL + loc;
            float l  = rg[0];
            float tt = rg[LL];
            float r  = rg[2 * LL];
            float bb = rg[3 * LL];
            x1 = fminf(fmaxf(lx - l,  0.f), wmax);
            y1 = fminf(fmaxf(ly - tt, 0.f), hmax);
            x2 = fminf(fmaxf(lx + r,  0.f), wmax);
            y2 = fminf(fmaxf(ly + bb, 0.f), hmax);
            float wsz = x2 - x1 + 1.0f;
            float hsz = y2 - y1 + 1.0f;
            val = (s > 0.f && wsz >= 0.f && hsz >= 0.f) ? 1.0f : 0.f;
            sc  = val * sqrtf(fmaxf(s, 0.f));
            x1 *= val; y1 *= val; x2 *= val; y2 *= val;
        }
        float* ob = out + ((size_t)i * KTOP + j) * 4;
        ob[0] = x1; ob[1] = y1; ob[2] = x2; ob[3] = y2;
        out[NIMG * KTOP * 4 + i * KTOP + j]                    = sc;   // scores
        out[NIMG * KTOP * 4 + NIMG * KTOP + i * KTOP + j]      = lab;  // labels
        out[NIMG * KTOP * 4 + 2 * NIMG * KTOP + i * KTOP + j]  = val;  // valid
    }
}

// ---------------- host launcher ----------------
extern "C" void kernel_launch(void* const* d_in, const int* in_sizes, int n_in,
                              void* d_out, int out_size, void* d_ws, size_t ws_size,
                              hipStream_t stream) {
    // input order: locations, box_cls, box_regression, centerness, image_sizes
    const float* box_cls = (const float*)d_in[1];
    const float* box_reg = (const float*)d_in[2];
    const float* ctr     = (const float*)d_in[3];
    const int*   imsz    = (const int*)d_in[4];
    float* out = (float*)d_out;

    char* ws = (char*)d_ws;
    unsigned long long* buf = (unsigned long long*)(ws);           // 524288 B
    float*    ctrsig = (float*)(ws + 524288);                      // 1638400 B
    unsigned* hist   = (unsigned*)(ws + 2162688);                  // 262144 B
    unsigned* Tbin   = (unsigned*)(ws + 2424832);                  // 64 B
    unsigned* cnt    = (unsigned*)(ws + 2424896);                  // 64 B

    k_init   <<<256, 256, 0, stream>>>(hist, cnt);
    k_ctr    <<<(NIMG * LL + 255) / 256, 256, 0, stream>>>(ctr, ctrsig, NIMG * LL);
    k_hist   <<<NIMG * NCLS, 256, 0, stream>>>(box_cls, ctrsig, hist);
    k_thresh <<<1, 64, 0, stream>>>(hist, Tbin);
    k_collect<<<NIMG * NCLS, 256, 0, stream>>>(box_cls, ctrsig, Tbin, cnt, buf);
    k_final  <<<NIMG, 1024, 0, stream>>>(cnt, buf, box_reg, imsz, out);
}